// TemporalPatternDecoupling_6614249636699
// MI455X (gfx1250) — compile-verified
//
#include <hip/hip_runtime.h>
#include <stdint.h>

#define ROWLEN 4096
#define FLEN   8
#define CALEN  ((ROWLEN + FLEN - 1) / 2)   // 2051

// db4 dec_lo taps
#define D0 (-0.010597401784997278f)
#define D1 ( 0.032883011666982945f)
#define D2 ( 0.030841381835986965f)
#define D3 (-0.18703481171888114f)
#define D4 (-0.02798376941698385f)
#define D5 ( 0.6308807679295904f)
#define D6 ( 0.7148465705525415f)
#define D7 ( 0.23037781330885523f)

typedef unsigned int u32x4 __attribute__((ext_vector_type(4)));
typedef int          i32x8 __attribute__((ext_vector_type(8)));
typedef int          i32x4 __attribute__((ext_vector_type(4)));
typedef float        f32x4 __attribute__((ext_vector_type(4)));

static __device__ __forceinline__ int sym_idx(int i) {
    // pywt 'symmetric' extension, single reflection is enough for pad=7
    if (i < 0)        i = -1 - i;
    if (i >= ROWLEN)  i = 2 * ROWLEN - 1 - i;
    return i;
}

__global__ __launch_bounds__(256)
void dwt_db4_lowhigh_kernel(const float* __restrict__ x,
                            float* __restrict__ low_out,
                            float* __restrict__ high_out) {
    __shared__ float xs[ROWLEN];       // 16 KB: one row
    __shared__ float ca[CALEN + 1];    // approx coeffs

    const int row = blockIdx.x;
    const int tid = threadIdx.x;
    const float* xrow = x + (size_t)row * ROWLEN;

    // ---- Stage 0: DMA the row into LDS via the Tensor Data Mover ----
#if defined(__AMDGCN__) && defined(__has_builtin)
#if __has_builtin(__builtin_amdgcn_tensor_load_to_lds) && __has_builtin(__builtin_amdgcn_s_wait_tensorcnt)
    if (tid < 32) {   // wave-uniform: only wave 0 issues the TDM op
        uint64_t gaddr = (uint64_t)(uintptr_t)xrow;
        uint32_t laddr = (uint32_t)(uintptr_t)&xs[0];   // low 32 bits = LDS byte addr
        // D# group 0: count=1 | lds_addr | global_addr[56:0] | type=2
        u32x4 g0 = { 1u,
                     laddr,
                     (uint32_t)(gaddr & 0xFFFFFFFFu),
                     (uint32_t)((gaddr >> 32) & 0x01FFFFFFu) | 0x80000000u };
        // D# group 1: data_size=4B(2), tensor_dim0=4096, tensor_dim1=1,
        //             tile_dim0=4096, tile_dim1=1, tensor_dim0_stride=4096
        i32x8 g1 = { 0x00020000,            // wg_mask=0, data_size=2 (4 bytes)
                     0x10000000,            // tensor_dim0[15:0]=4096 in bits[31:16]
                     0x00010000,            // tensor_dim1=1 in bits[31:16]
                     0x10000000,            // tile_dim0=4096 in bits[31:16]
                     0x00000001,            // tile_dim1=1, tile_dim2=0
                     0x00001000,            // tensor_dim0_stride=4096
                     0, 0 };
        i32x4 gz4 = { 0, 0, 0, 0 };         // groups 2/3 unused (2D tensor)
        i32x8 gz8 = { 0, 0, 0, 0, 0, 0, 0, 0 };
        __builtin_amdgcn_tensor_load_to_lds(g0, g1, gz4, gz4, gz8, 0);
        __builtin_amdgcn_s_wait_tensorcnt(0);
    }
#else
    for (int i = tid; i < ROWLEN; i += 256) xs[i] = xrow[i];
#endif
#else
    for (int i = tid; i < ROWLEN; i += 256) xs[i] = xrow[i];
#endif
    __syncthreads();

    // ---- Stage 1: analysis. ca[j] = sum_t dec_lo[7-t] * xs[sym(2j-6+t)] ----
    for (int j = tid; j < CALEN; j += 256) {
        const int b = 2 * j - 6;
        float acc;
        if (j >= 3 && j <= 2047) {          // fully interior window
            acc = D7 * xs[b]     + D6 * xs[b + 1] + D5 * xs[b + 2] + D4 * xs[b + 3]
                + D3 * xs[b + 4] + D2 * xs[b + 5] + D1 * xs[b + 6] + D0 * xs[b + 7];
        } else {                             // mirrored boundary
            acc = D7 * xs[sym_idx(b)]     + D6 * xs[sym_idx(b + 1)]
                + D5 * xs[sym_idx(b + 2)] + D4 * xs[sym_idx(b + 3)]
                + D3 * xs[sym_idx(b + 4)] + D2 * xs[sym_idx(b + 5)]
                + D1 * xs[sym_idx(b + 6)] + D0 * xs[sym_idx(b + 7)];
        }
        ca[j] = acc;
    }
    __syncthreads();

    // ---- Stage 2: synthesis (details zeroed) + residual; NT 128-bit stores ----
    float* __restrict__ lrow = low_out  + (size_t)row * ROWLEN;
    float* __restrict__ hrow = high_out + (size_t)row * ROWLEN;
    for (int g = tid; g < ROWLEN / 4; g += 256) {
        const int n = 4 * g;
        const int j = 2 * g;
        const float c0 = ca[j],     c1 = ca[j + 1], c2 = ca[j + 2];
        const float c3 = ca[j + 3], c4 = ca[j + 4];
        f32x4 lo;
        lo.x = D1 * c0 + D3 * c1 + D5 * c2 + D7 * c3;   // n   (even)
        lo.y = D0 * c0 + D2 * c1 + D4 * c2 + D6 * c3;   // n+1 (odd)
        lo.z = D1 * c1 + D3 * c2 + D5 * c3 + D7 * c4;   // n+2 (even)
        lo.w = D0 * c1 + D2 * c2 + D4 * c3 + D6 * c4;   // n+3 (odd)
        const f32x4 xv = *(const f32x4*)&xs[n];
        f32x4 hi = xv - lo;
        // Outputs are 256 MB of write-once stream (> 192 MB L2): non-temporal
        __builtin_nontemporal_store(lo, (f32x4*)&lrow[n]);
        __builtin_nontemporal_store(hi, (f32x4*)&hrow[n]);
    }
}

extern "C" void kernel_launch(void* const* d_in, const int* in_sizes, int n_in,
                              void* d_out, int out_size, void* d_ws, size_t ws_size,
                              hipStream_t stream) {
    (void)n_in; (void)out_size; (void)d_ws; (void)ws_size;
    const float* x = (const float*)d_in[0];
    const int N    = in_sizes[0];          // 16*512*4096 = 33,554,432
    const int rows = N / ROWLEN;           // 8192
    float* low  = (float*)d_out;           // outputs concatenated: (low, high)
    float* high = low + (size_t)N;
    dwt_db4_lowhigh_kernel<<<rows, 256, 0, stream>>>(x, low, high);
}